// NNDModule_33844342293198
// MI455X (gfx1250) — compile-verified
//
#include <hip/hip_runtime.h>

// Chamfer / NN squared distance for MI455X (gfx1250, wave32).
// V_WMMA_F32_16X16X4_F32 with the spare K3 slot carrying the ||y||^2 term:
//   A: K0..K2 = -2*x, K3 = 1      (16x4 per fragment, two fragments = 32 rows/wave)
//   B: K0..K2 =    y, K3 = ||y||^2
// => C[m,n] = ||y_n||^2 - 2*x_m.y_n  directly; running min + final ||x||^2 add.

typedef float v2f __attribute__((ext_vector_type(2)));
typedef float v8f __attribute__((ext_vector_type(8)));

__global__ __launch_bounds__(256) void nnd_min_kernel(
    const float* __restrict__ q,   // [B, Nq, 3] query points (rows of A)
    const float* __restrict__ r,   // [B, Nr, 3] reference points (cols of B)
    float* __restrict__ out,       // [B, Nq] min squared distance per query
    int Nq, int Nr)
{
  const int lane  = threadIdx.x & 31;
  const int wib   = threadIdx.x >> 5;
  const int wpb   = blockDim.x >> 5;
  const int gwave = blockIdx.x * wpb + wib;

  const int tilesPerBatch = Nq >> 5;        // 32 query rows per wave
  const int b    = gwave / tilesPerBatch;
  const int tile = gwave % tilesPerBatch;
  const int row0 = tile << 5;

  const float* qb = q + (size_t)b * Nq * 3;
  const float* rb = r + (size_t)b * Nr * 3;

  const int lrow = lane & 15;   // row (A) / col (B) index within 16
  const int hh   = lane >> 4;   // half-wave: 0 -> K0,K1 ; 1 -> K2,K3

  // ---- A fragments: two 16-row query tiles, scaled by -2, K3 = 1.0 ----
  const float* qp0 = qb + (size_t)(row0 + lrow) * 3;
  const float* qp1 = qb + (size_t)(row0 + 16 + lrow) * 3;
  const float q0x = qp0[0], q0y = qp0[1], q0z = qp0[2];
  const float q1x = qp1[0], q1y = qp1[1], q1z = qp1[2];
  const float q20 = q0x * q0x + q0y * q0y + q0z * q0z;
  const float q21 = q1x * q1x + q1y * q1y + q1z * q1z;

  v2f a0, a1;
  a0.x = hh ? (-2.0f * q0z) : (-2.0f * q0x);   // VGPR0: K0 / K2
  a0.y = hh ? 1.0f          : (-2.0f * q0y);   // VGPR1: K1 / K3
  a1.x = hh ? (-2.0f * q1z) : (-2.0f * q1x);
  a1.y = hh ? 1.0f          : (-2.0f * q1y);

  // Per-lane ||q||^2 for the 8 rows each C VGPR covers (M = i + 8*hh).
  float s0[8], s1[8];
  #pragma unroll
  for (int i = 0; i < 8; ++i) {
    s0[i] = __shfl(q20, i + 8 * hh, 32);
    s1[i] = __shfl(q21, i + 8 * hh, 32);
  }

  float rm0[8], rm1[8];
  #pragma unroll
  for (int i = 0; i < 8; ++i) { rm0[i] = 3.0e38f; rm1[i] = 3.0e38f; }

  // ---- sweep reference points in 16-wide tiles ----
  for (int m0 = 0; m0 < Nr; m0 += 16) {
    const float* rp = rb + (size_t)(m0 + lrow) * 3;
    const float rx = rp[0], ry = rp[1], rz = rp[2];
    const float r2 = rx * rx + ry * ry + rz * rz;   // ||y||^2 for col = lrow

    v2f bf;
    bf.x = hh ? rz : rx;          // K0 / K2
    bf.y = hh ? r2 : ry;          // K1 / K3  (K3 carries ||y||^2)

    v8f c0 = {}, c1 = {};
    c0 = __builtin_amdgcn_wmma_f32_16x16x4_f32(
        false, a0, false, bf, (short)0, c0, false, false);
    c1 = __builtin_amdgcn_wmma_f32_16x16x4_f32(
        false, a1, false, bf, (short)0, c1, false, false);

    #pragma unroll
    for (int i = 0; i < 8; ++i) {
      rm0[i] = fminf(rm0[i], c0[i]);
      rm1[i] = fminf(rm1[i], c1[i]);
    }
  }

  // ---- min across the 16 lanes of each half (cols of each row) ----
  #pragma unroll
  for (int mask = 1; mask <= 8; mask <<= 1) {
    #pragma unroll
    for (int i = 0; i < 8; ++i) {
      rm0[i] = fminf(rm0[i], __shfl_xor(rm0[i], mask, 32));
      rm1[i] = fminf(rm1[i], __shfl_xor(rm1[i], mask, 32));
    }
  }

  // rows row0 + i + 8*hh (frag0) and row0 + 16 + i + 8*hh (frag1)
  if (lrow == 0) {
    float* ob0 = out + (size_t)b * Nq + row0 + 8 * hh;
    float* ob1 = ob0 + 16;
    #pragma unroll
    for (int i = 0; i < 8; ++i) {
      ob0[i] = rm0[i] + s0[i];
      ob1[i] = rm1[i] + s1[i];
    }
  }
}

extern "C" void kernel_launch(void* const* d_in, const int* in_sizes, int n_in,
                              void* d_out, int out_size, void* d_ws, size_t ws_size,
                              hipStream_t stream) {
  (void)n_in; (void)d_ws; (void)ws_size; (void)out_size;
  const float* x = (const float*)d_in[0];   // [B, N, 3]
  const float* y = (const float*)d_in[1];   // [B, M, 3]
  float* out = (float*)d_out;               // dist1 [B,N] ++ dist2 [B,M]

  const int B = 8;
  const int N = in_sizes[0] / (B * 3);      // 8192
  const int M = in_sizes[1] / (B * 3);      // 8192

  const int threads = 256;                  // 8 wave32 per block
  const int wavesPerBlock = threads / 32;

  // dist1: queries = x, refs = y
  {
    int waves = B * (N / 32);
    nnd_min_kernel<<<dim3(waves / wavesPerBlock), dim3(threads), 0, stream>>>(
        x, y, out, N, M);
  }
  // dist2: queries = y, refs = x
  {
    int waves = B * (M / 32);
    nnd_min_kernel<<<dim3(waves / wavesPerBlock), dim3(threads), 0, stream>>>(
        y, x, out + (size_t)B * N, M, N);
  }
}